// NFFT_29781303230647
// MI455X (gfx1250) — compile-verified
//
#include <hip/hip_runtime.h>
#include <math.h>

// ---------------------------------------------------------------------------
// NFFT (type 2): deconv+pad -> 2D DFT via WMMA matmul (W * G * W^T) -> gather
// Grids: N=256x256, oversampled n=512x512, m=4 (8x8 taps), B=2, C=2, M=200000
// fftshift/ifftshift folded into (-1)^(r+c) signs at input/output of the DFT.
// ---------------------------------------------------------------------------

typedef float v2f __attribute__((ext_vector_type(2)));
typedef float v8f __attribute__((ext_vector_type(8)));

#define PI_F  3.14159265358979323846f
#define BKB   4.71238898038468985769f   /* (2 - 1/sigma)*pi, sigma=2 -> 1.5*pi */
#define NOS   512
#define NIMG  4                          /* B*C */
#define PLANE (NOS*NOS)                  /* 262144 */

// ------------------------- DFT twiddle matrices ----------------------------
// W[k][n] = exp(-2*pi*i*k*n/512) = Wc + i*Ws  (Ws stores -sin)
__global__ __launch_bounds__(256) void nfft_genW(float* __restrict__ Wc,
                                                 float* __restrict__ Ws) {
    int tid = blockIdx.x * blockDim.x + threadIdx.x;
    if (tid >= PLANE) return;
    int k = tid >> 9, n = tid & 511;
    int kn = (k * n) & 511;                       // exact mod-512 phase
    float th = (2.0f * PI_F / 512.0f) * (float)kn;
    float s, c;
    sincosf(th, &s, &c);
    Wc[tid] = c;
    Ws[tid] = -s;
}

// ------------------------- deconvolve + pad + shift ------------------------
__device__ __forceinline__ float i0_asym(float z) {
    // I0(z) ~ e^z/sqrt(2*pi*z)*(1 + 1/(8z) + 9/(128z^2) + 75/(1024z^3) + ...)
    float iz = 1.0f / z;
    float p = 1.0f + iz * (0.125f + iz * (0.0703125f +
              iz * (0.0732421875f + iz * 0.112152099609375f)));
    return expf(z) * rsqrtf(2.0f * PI_F * z) * p;
}
__device__ __forceinline__ float phihat(int ind) {
    float tpk = (2.0f * PI_F / 512.0f) * (float)ind;
    return i0_asym(4.0f * sqrtf(BKB * BKB - tpk * tpk));
}

__global__ __launch_bounds__(256) void nfft_prep(const float* __restrict__ fr,
                                                 const float* __restrict__ fi,
                                                 float* __restrict__ Ar,
                                                 float* __restrict__ Ai) {
    int tid = blockIdx.x * blockDim.x + threadIdx.x;
    if (tid >= NIMG * PLANE) return;
    int img = tid >> 18;
    int rem = tid & (PLANE - 1);
    int r = rem >> 9, c = rem & 511;
    float vr = 0.0f, vi = 0.0f;
    if (r >= 128 && r < 384 && c >= 128 && c < 384) {
        int i = r - 128, j = c - 128;
        float ph = phihat(i - 128) * phihat(j - 128);
        float s = ((r + c) & 1) ? -1.0f : 1.0f;   // fftshift fold on input
        float inv = s / ph;
        int off = img * 65536 + i * 256 + j;
        vr = fr[off] * inv;
        vi = fi[off] * inv;
    }
    Ar[tid] = vr;
    Ai[tid] = vi;
}

// ------------------------- WMMA complex DFT pass ---------------------------
// Computes U = W * X (XTRANS=0: X[k][j]; XTRANS=1: X^T, i.e. X[j][k]).
// OMODE 0: planar store U[i][j]. OMODE 1: store transposed with ifftshift
// sign into interleaved float4 grid (c0.re,c0.im,c1.re,c1.im) per pixel.
template <int XTRANS, int OMODE>
__global__ __launch_bounds__(256) void nfft_dft_pass(
    const float* __restrict__ Wc, const float* __restrict__ Ws,
    const float* __restrict__ Xr, const float* __restrict__ Xi,
    float* __restrict__ OutA, float* __restrict__ OutB) {
    const int lane   = threadIdx.x;                      // wave32
    const int waveId = blockIdx.x * blockDim.y + threadIdx.y;
    const int img = waveId >> 10;                        // 1024 tiles per image
    const int t   = waveId & 1023;
    const int tm  = t >> 5, tn = t & 31;
    const int lmod = lane & 15, lhi = lane >> 4;
    const int rowA = tm * 16 + lmod;                     // A-matrix M index
    const int col  = tn * 16 + lmod;                     // B/C N index
    const int kb   = lhi * 2;                            // per-lane K base
    const float* xr = Xr + img * PLANE;
    const float* xi = Xi + img * PLANE;

    v8f acc_rr = {}, acc_si = {}, acc_ci = {}, acc_sr = {};
    for (int k0 = 0; k0 < 512; k0 += 4) {
        // A (16x4 f32): lane L -> A[L%16][2*(L/16)+{0,1}]
        v2f ac = *(const v2f*)(Wc + rowA * 512 + k0 + kb);
        v2f as = *(const v2f*)(Ws + rowA * 512 + k0 + kb);
        v2f br, bi;
        if (XTRANS) {
            br = *(const v2f*)(xr + col * 512 + k0 + kb);
            bi = *(const v2f*)(xi + col * 512 + k0 + kb);
        } else {
            br.x = xr[(k0 + kb) * 512 + col];
            br.y = xr[(k0 + kb + 1) * 512 + col];
            bi.x = xi[(k0 + kb) * 512 + col];
            bi.y = xi[(k0 + kb + 1) * 512 + col];
        }
        acc_rr = __builtin_amdgcn_wmma_f32_16x16x4_f32(false, ac, false, br,
                                                       (short)0, acc_rr, false, false);
        acc_si = __builtin_amdgcn_wmma_f32_16x16x4_f32(false, as, false, bi,
                                                       (short)0, acc_si, false, false);
        acc_ci = __builtin_amdgcn_wmma_f32_16x16x4_f32(false, ac, false, bi,
                                                       (short)0, acc_ci, false, false);
        acc_sr = __builtin_amdgcn_wmma_f32_16x16x4_f32(false, as, false, br,
                                                       (short)0, acc_sr, false, false);
    }
    v8f re = acc_rr - acc_si;   // (Wc + i*Ws)(Xr + i*Xi): real part
    v8f im = acc_ci + acc_sr;   // imag part

    if (OMODE == 0) {
        float* orp = OutA + img * PLANE;
        float* oip = OutB + img * PLANE;
#pragma unroll
        for (int v = 0; v < 8; ++v) {
            int row = tm * 16 + v + 8 * lhi;             // C/D: M = v + 8*(L/16)
            orp[row * 512 + col] = re[v];
            oip[row * 512 + col] = im[v];
        }
    } else {
        int b = img >> 1, ch = img & 1;
        float* gf = OutA + b * (PLANE * 4);
#pragma unroll
        for (int v = 0; v < 8; ++v) {
            int row = tm * 16 + v + 8 * lhi;
            int pix = col * 512 + row;                   // transposed store
            float s = ((col + row) & 1) ? -1.0f : 1.0f;  // ifftshift fold
            float2 val = make_float2(s * re[v], s * im[v]);
            *(float2*)(gf + pix * 4 + ch * 2) = val;
        }
    }
}

// ------------------------- windowed gather ---------------------------------
__global__ __launch_bounds__(256) void nfft_gather(const float* __restrict__ x,
                                                   const float* __restrict__ Gf,
                                                   float* __restrict__ out,
                                                   int M) {
    int tid = blockIdx.x * blockDim.x + threadIdx.x;
    if (tid >= 2 * M) return;
    int b = tid / M, p = tid - b * M;
    float x0 = x[(b * M + p) * 2 + 0];
    float x1 = x[(b * M + p) * 2 + 1];
    int base0 = (int)ceilf(x0 * 512.0f) - 4;
    int base1 = (int)ceilf(x1 * 512.0f) - 4;

    float wx[8], wy[8];
    int ix[8], iy[8];
#pragma unroll
    for (int i = 0; i < 8; ++i) {
        float u = x0 * 512.0f - (float)(base0 + i);
        float tt = 16.0f - u * u;
        float w = 0.0f;
        if (tt > 0.0f) {
            float sq = sqrtf(tt);
            float e = expf(BKB * sq);
            w = 0.5f * (e - 1.0f / e) / (sq * PI_F);     // sinh(b*sq)/(sq*pi)
        }
        wx[i] = w;
        ix[i] = (base0 + i + 768) & 511;                 // wrap onto grid
        u = x1 * 512.0f - (float)(base1 + i);
        tt = 16.0f - u * u;
        w = 0.0f;
        if (tt > 0.0f) {
            float sq = sqrtf(tt);
            float e = expf(BKB * sq);
            w = 0.5f * (e - 1.0f / e) / (sq * PI_F);
        }
        wy[i] = w;
        iy[i] = (base1 + i + 768) & 511;
    }

    const float* g = Gf + b * (PLANE * 4);
    float a0r = 0.f, a0i = 0.f, a1r = 0.f, a1i = 0.f;
#pragma unroll
    for (int i = 0; i < 8; ++i) {
        float wxi = wx[i];
        const float* grow = g + ix[i] * 2048;            // 512 pixels * 4 floats
#pragma unroll
        for (int j = 0; j < 8; ++j) {
            float w = wxi * wy[j];
            float4 v = *(const float4*)(grow + iy[j] * 4);
            a0r += w * v.x; a0i += w * v.y;
            a1r += w * v.z; a1i += w * v.w;
        }
    }
    // output (B,C,M) complex64, interleaved re/im
    out[((b * 2 + 0) * M + p) * 2 + 0] = a0r;
    out[((b * 2 + 0) * M + p) * 2 + 1] = a0i;
    out[((b * 2 + 1) * M + p) * 2 + 0] = a1r;
    out[((b * 2 + 1) * M + p) * 2 + 1] = a1i;
}

// ------------------------- launch ------------------------------------------
extern "C" void kernel_launch(void* const* d_in, const int* in_sizes, int n_in,
                              void* d_out, int out_size, void* d_ws, size_t ws_size,
                              hipStream_t stream) {
    const float* xin = (const float*)d_in[0];   // (B,1,M,2) f32
    const float* fhr = (const float*)d_in[1];   // (B,C,256,256) f32
    const float* fhi = (const float*)d_in[2];   // (B,C,256,256) f32
    const int M = in_sizes[0] / 4;              // B=2, last dim 2

    float* ws = (float*)d_ws;
    float* Wc  = ws;                            // 262144
    float* Wsm = Wc + PLANE;                    // 262144
    float* Ar  = Wsm + PLANE;                   // 4 planes
    float* Ai  = Ar + NIMG * PLANE;
    float* Tr  = Ai + NIMG * PLANE;
    float* Ti  = Tr + NIMG * PLANE;
    float* Gf  = Ti + NIMG * PLANE;             // 2 * 512*512*4 floats interleaved

    nfft_genW<<<(PLANE + 255) / 256, 256, 0, stream>>>(Wc, Wsm);
    nfft_prep<<<(NIMG * PLANE + 255) / 256, 256, 0, stream>>>(fhr, fhi, Ar, Ai);

    dim3 mb(32, 8);                              // 8 wave32 per block
    // pass 1: T = W * A        (transform axis 0)
    nfft_dft_pass<0, 0><<<(NIMG * 1024) / 8, mb, 0, stream>>>(Wc, Wsm, Ar, Ai, Tr, Ti);
    // pass 2: F = (W * T^T)^T  (transform axis 1), ifftshift sign, interleave
    nfft_dft_pass<1, 1><<<(NIMG * 1024) / 8, mb, 0, stream>>>(Wc, Wsm, Tr, Ti, Gf, nullptr);

    nfft_gather<<<(2 * M + 255) / 256, 256, 0, stream>>>(xin, Gf, (float*)d_out, M);
}